// SequenceMemoryUpdater_71786083385644
// MI455X (gfx1250) — compile-verified
//
#include <hip/hip_runtime.h>

// ---------------------------------------------------------------------------
// TGN memory updater for MI455X (gfx1250, wave32).
//   updated_memory  = memory.at[node_ids].set(GRUCell(messages, memory[node_ids]))
//   updated_last_up = last_update.at[node_ids].set(timestamps)
//
// f32 GEMMs on V_WMMA_F32_16X16X4_F32. Block = 1024 threads (32 waves),
// M-tile = 64 update rows. Each wave owns 16 gate columns and 4 M-subtiles:
// per K-step -> 3 global_load_b64 (B frags) + 4 LDS A frags + 12 WMMAs,
// i.e. 4 WMMA per L2 weight load (vs 1:1 with a 16-row tile).
// ---------------------------------------------------------------------------

typedef __attribute__((ext_vector_type(2))) float v2f;
typedef __attribute__((ext_vector_type(8))) float v8f;

#define N_NODES 262144
#define MEM_DIM 512
#define MSG_DIM 1024
#define N_UPD   65536
#define TM      64           // update rows per block
#define KC      512          // K chunk resident in LDS
#define NSUB    (TM / 16)    // 4 M-subtiles per wave

__device__ __forceinline__ float fast_sigmoid(float x) {
    return 1.0f / (1.0f + __expf(-x));
}
__device__ __forceinline__ float fast_tanh(float x) {
    return 1.0f - 2.0f / (__expf(2.0f * x) + 1.0f);
}

// One K=512 GEMM chunk: A from LDS (row stride KC), B streamed from L2.
// ISA fragment layout (wave32): lane L -> n = L&15, kset = (L>>4)*2.
//   A: lane holds rows m = subtile*16 + n, elements (k+kset, k+kset+1)
//   B: lane holds col n,                 elements (k+kset, k+kset+1)
__device__ __forceinline__ void mma_chunk(const float* __restrict__ Abase,
                                          const float* __restrict__ Br,
                                          const float* __restrict__ Bz,
                                          const float* __restrict__ Bn,
                                          v8f* accR, v8f* accZ, v8f* accN)
{
    #pragma unroll 2
    for (int k = 0; k < KC; k += 4) {
        v2f br = *(const v2f*)(Br + k);
        v2f bz = *(const v2f*)(Bz + k);
        v2f bn = *(const v2f*)(Bn + k);
        #pragma unroll
        for (int s = 0; s < NSUB; ++s) {
            v2f a = *(const v2f*)(Abase + s * (16 * KC) + k);
            accR[s] = __builtin_amdgcn_wmma_f32_16x16x4_f32(false, a, false, br, (short)0, accR[s], false, false);
            accZ[s] = __builtin_amdgcn_wmma_f32_16x16x4_f32(false, a, false, bz, (short)0, accZ[s], false, false);
            accN[s] = __builtin_amdgcn_wmma_f32_16x16x4_f32(false, a, false, bn, (short)0, accN[s], false, false);
        }
    }
}

extern "C" __global__ __launch_bounds__(1024)
void tgn_gru_wmma_kernel(const float* __restrict__ memory,
                         const int*   __restrict__ node_ids,
                         const float* __restrict__ messages,
                         const float* __restrict__ timestamps,
                         const float* __restrict__ W_ih,   // [1536,1024] row-major
                         const float* __restrict__ W_hh,   // [1536, 512] row-major
                         const float* __restrict__ b_ih,   // [1536]
                         const float* __restrict__ b_hh,   // [1536]
                         float* __restrict__ out_mem,      // [262144,512]
                         float* __restrict__ out_lu)       // [262144]
{
    extern __shared__ float smem[];
    float* Xc  = smem;                    // [64][512] message K-chunk  (128 KB)
    float* Hl  = smem + TM * KC;          // [64][512] gathered memory  (128 KB)
    int*   nid = (int*)(smem + 2 * TM * KC); // [64] node ids

    const int tid = threadIdx.x;
    const int e0  = blockIdx.x * TM;

    // ---- Phase 1: stage H tile (gather), X chunk 0, node ids --------------
    if (tid < TM) {
        int node = node_ids[e0 + tid];
        nid[tid] = node;
        out_lu[node] = timestamps[e0 + tid];   // scatter timestamps
    }
    {   // H tile: 64*512 floats = 8192 float4, 1024 threads -> 8 each
        float4* dst = (float4*)Hl;
        #pragma unroll
        for (int i = 0; i < 8; ++i) {
            int linear = tid + i * 1024;
            int m    = linear >> 7;            // row 0..63
            int c4   = linear & 127;
            int node = node_ids[e0 + m];
            dst[linear] = ((const float4*)memory)[(size_t)node * (MEM_DIM / 4) + c4];
        }
    }
    {   // X chunk 0: columns [0,512)
        float4* dst = (float4*)Xc;
        #pragma unroll
        for (int i = 0; i < 8; ++i) {
            int linear = tid + i * 1024;
            int m  = linear >> 7;
            int c4 = linear & 127;
            dst[linear] = ((const float4*)messages)[(size_t)(e0 + m) * (MSG_DIM / 4) + c4];
        }
    }
    __syncthreads();

    // ---- Phase 2: WMMA accumulation ---------------------------------------
    const int lane = tid & 31;
    const int wave = tid >> 5;                 // 0..31
    const int j0   = wave * 16;                // gate columns [j0, j0+16)
    const int n    = lane & 15;
    const int kset = (lane >> 4) << 1;         // 0 or 2

    v8f accR[NSUB]  = {};
    v8f accZ[NSUB]  = {};
    v8f accIN[NSUB] = {};
    v8f accHN[NSUB] = {};

    const float* Wr_i = W_ih + (size_t)(       j0 + n) * MSG_DIM + kset;
    const float* Wz_i = W_ih + (size_t)( 512 + j0 + n) * MSG_DIM + kset;
    const float* Wn_i = W_ih + (size_t)(1024 + j0 + n) * MSG_DIM + kset;
    const float* Wr_h = W_hh + (size_t)(       j0 + n) * MEM_DIM + kset;
    const float* Wz_h = W_hh + (size_t)( 512 + j0 + n) * MEM_DIM + kset;
    const float* Wn_h = W_hh + (size_t)(1024 + j0 + n) * MEM_DIM + kset;

    const float* Ax = Xc + n * KC + kset;      // A base (subtile stride 16*KC)
    const float* Ah = Hl + n * KC + kset;

    // gi, K chunk 0
    mma_chunk(Ax, Wr_i, Wz_i, Wn_i, accR, accZ, accIN);

    // swap in X chunk 1 (accumulators live in registers across the swap)
    __syncthreads();
    {
        float4* dst = (float4*)Xc;
        #pragma unroll
        for (int i = 0; i < 8; ++i) {
            int linear = tid + i * 1024;
            int m  = linear >> 7;
            int c4 = linear & 127;
            dst[linear] = ((const float4*)messages)[(size_t)(e0 + m) * (MSG_DIM / 4) + (KC / 4) + c4];
        }
    }
    __syncthreads();

    // gi, K chunk 1
    mma_chunk(Ax, Wr_i + KC, Wz_i + KC, Wn_i + KC, accR, accZ, accIN);

    // gh (K = 512, H tile resident; h_n kept separate for the tanh gate)
    mma_chunk(Ah, Wr_h, Wz_h, Wn_h, accR, accZ, accHN);

    // ---- Phase 3: gate nonlinearity + scatter -----------------------------
    const float bias_r  = b_ih[       j0 + n] + b_hh[       j0 + n];
    const float bias_z  = b_ih[ 512 + j0 + n] + b_hh[ 512 + j0 + n];
    const float bias_in = b_ih[1024 + j0 + n];
    const float bias_hn = b_hh[1024 + j0 + n];

    const int mbase = (lane >> 4) << 3;        // C/D layout: m = v + (lane>=16 ? 8 : 0)
    #pragma unroll
    for (int s = 0; s < NSUB; ++s) {
        #pragma unroll
        for (int v = 0; v < 8; ++v) {
            int   m  = s * 16 + mbase + v;
            float r  = fast_sigmoid(accR[s][v] + bias_r);
            float z  = fast_sigmoid(accZ[s][v] + bias_z);
            float nn = fast_tanh(accIN[s][v] + bias_in + r * (accHN[s][v] + bias_hn));
            float h  = Hl[m * KC + j0 + n];
            float hnew = (1.0f - z) * nn + z * h;
            out_mem[(size_t)nid[m] * MEM_DIM + j0 + n] = hnew;
        }
    }
}

extern "C" void kernel_launch(void* const* d_in, const int* in_sizes, int n_in,
                              void* d_out, int out_size, void* d_ws, size_t ws_size,
                              hipStream_t stream) {
    (void)in_sizes; (void)n_in; (void)out_size; (void)d_ws; (void)ws_size;

    const float* memory      = (const float*)d_in[0];
    const float* last_update = (const float*)d_in[1];
    const int*   node_ids    = (const int*)  d_in[2];
    const float* messages    = (const float*)d_in[3];
    const float* timestamps  = (const float*)d_in[4];
    const float* W_ih        = (const float*)d_in[5];
    const float* W_hh        = (const float*)d_in[6];
    const float* b_ih        = (const float*)d_in[7];
    const float* b_hh        = (const float*)d_in[8];

    float* out_mem = (float*)d_out;
    float* out_lu  = out_mem + (size_t)N_NODES * MEM_DIM;

    // Clone untouched state first (stream-ordered before the update kernel).
    hipMemcpyAsync(out_mem, memory, (size_t)N_NODES * MEM_DIM * sizeof(float),
                   hipMemcpyDeviceToDevice, stream);
    hipMemcpyAsync(out_lu, last_update, (size_t)N_NODES * sizeof(float),
                   hipMemcpyDeviceToDevice, stream);

    dim3 grid(N_UPD / TM);          // 1024 blocks
    dim3 block(1024);               // 32 waves, one per 16-column gate block
    size_t lds_bytes = (size_t)(2 * TM * KC) * sizeof(float) + TM * sizeof(int); // 256 KB + ids
    tgn_gru_wmma_kernel<<<grid, block, lds_bytes, stream>>>(
        memory, node_ids, messages, timestamps, W_ih, W_hh, b_ih, b_hh,
        out_mem, out_lu);
}